// TGCNGraphConvolution_46746424050384
// MI455X (gfx1250) — compile-verified
//
#include <hip/hip_runtime.h>

// TGCN graph convolution, fused two-stage WMMA f32 kernel for gfx1250 (wave32).
// Stage 1: ax[b, m, :] = L[m, :] @ xh[b]   (xh = permuted concat: h cols 0..127, x cols 128..129)
// Stage 2: out[b, m, j] = ax[b, m, :130] @ Wperm[:130, j] + bias[j]
// B=128, N=400, K1=400, GC=130, DOUT=128.

typedef float v2f __attribute__((ext_vector_type(2)));
typedef float v4f __attribute__((ext_vector_type(4)));
typedef float v8f __attribute__((ext_vector_type(8)));

#define NNODE   400
#define BATCH   128
#define DOUT    128
#define KCHUNK  80     // K-chunk staged in LDS (400 = 5 * 80)
#define NCHUNK  5
#define XP      136    // xh_lds pitch (2*136 % 64 == 16 -> halves conflict-free)
#define LP      84     // L_lds  pitch (lane*84 % 64 distinct, halves disjoint)
#define AP      132    // ax_lds pitch (lane*132 % 64 = lane*4, halves disjoint)

// Permuted weight row k: k<128 -> h channel (orig gc=k+2), k in {128,129} -> x channel.
__device__ __forceinline__ float load_w(const float* __restrict__ w, int k, int j) {
    if (k >= 130) return 0.0f;                 // zero-pad rows 130..131
    const int g = (k < 128) ? ((k >> 1) + 1) : 0;
    const int c = k & 1;
    const int d = j >> 1, o = j & 1;
    return w[g * 256 + d * 4 + c * 2 + o];     // weights strides (256,4,2,1)
}

__global__ __launch_bounds__(256)
void tgcn_fused_wmma(const float* __restrict__ x,
                     const float* __restrict__ h,
                     const float* __restrict__ L,
                     const float* __restrict__ w,
                     const float* __restrict__ bias,
                     float* __restrict__ out) {
    __shared__ float xh_lds[KCHUNK * XP];      // 42.5 KB
    __shared__ float L_lds[16 * LP];           //  5.3 KB
    __shared__ float ax_lds[16 * AP];          //  8.3 KB

    const int b    = blockIdx.y;
    const int m0   = blockIdx.x * 16;
    const int tid  = threadIdx.x;
    const int wave = tid >> 5;                 // 8 wave32 waves
    const int lane = tid & 31;
    const int half = lane >> 4;                // selects K pair within WMMA operand
    const int l16  = lane & 15;

    v8f   acc  = {};                           // per-wave 16x16 WMMA accumulator
    float sacc = 0.0f;                         // wave 7: x-column scalar accumulator
    const int col = wave * 16 + l16;           // B column (h channel) for this lane

    for (int chunk = 0; chunk < NCHUNK; ++chunk) {
        const int k0 = chunk * KCHUNK;
        __syncthreads();                       // previous chunk fully consumed

        // ---- cooperative staging (all loads coalesced) ----
        const float* hsrc = h + (size_t)(b * NNODE + k0) * 128;
        for (int q = tid; q < KCHUNK * 32; q += 256) {        // 10 iters: b128 copies
            const int r = q >> 5, c4 = (q & 31) << 2;
            *(v4f*)&xh_lds[r * XP + c4] = *(const v4f*)&hsrc[r * 128 + c4];
        }
        const float* xsrc = x + (size_t)(b * NNODE + k0) * 2;
        if (tid < KCHUNK * 2)                                 // x -> cols 128..129
            xh_lds[(tid >> 1) * XP + 128 + (tid & 1)] = xsrc[tid];
        const float* Lsrc = L + (size_t)(m0 + (tid >> 4)) * NNODE + k0;
        for (int c = (tid & 15); c < KCHUNK; c += 16)         // 5 iters per thread
            L_lds[(tid >> 4) * LP + c] = Lsrc[c];

        if (chunk + 1 < NCHUNK) {                             // prefetch next h chunk
            const float* nxt = h + (size_t)(b * NNODE + k0 + KCHUNK) * 128;
            __builtin_prefetch(nxt + tid * 32, 0, 0);         // global_prefetch_b8
        }
        __syncthreads();

        // ---- stage 1 WMMA: A = L tile rows (LDS), B = xh chunk (LDS) ----
#pragma unroll 4
        for (int k = 0; k < KCHUNK; k += 4) {
            const int kk = k + 2 * half;
            v2f a = *(const v2f*)&L_lds[l16 * LP + kk];       // ds_load_b64
            v2f bb;
            bb.x = xh_lds[kk * XP + col];
            bb.y = xh_lds[(kk + 1) * XP + col];
            acc = __builtin_amdgcn_wmma_f32_16x16x4_f32(
                false, a, false, bb, (short)0, acc, false, false);
        }
        if (wave == 7) {                       // x columns: 16 rows x 2 cols = 32 lanes
#pragma unroll 4
            for (int k = 0; k < KCHUNK; ++k)
                sacc = fmaf(L_lds[l16 * LP + k], xh_lds[k * XP + 128 + half], sacc);
        }
    }

    // ---- spill ax (16 x 132) to LDS ----
#pragma unroll
    for (int i = 0; i < 8; ++i)                // D layout: VGPR i -> row i + 8*half
        ax_lds[(i + 8 * half) * AP + col] = acc[i];
    if (wave == 7) {
        ax_lds[l16 * AP + 128 + half] = sacc;  // x columns
        ax_lds[l16 * AP + 130 + half] = 0.0f;  // zero-pad cols 130..131
    }
    __syncthreads();

    // ---- stage 2 WMMA: out tile = ax[16 x 130] @ Wperm[130 x 128] ----
    const int j = wave * 16 + l16;             // output column (d*2+o)
    v8f acc2 = {};
    for (int k = 0; k < 132; k += 4) {
        const int kk = k + 2 * half;
        v2f a = *(const v2f*)&ax_lds[l16 * AP + kk];          // ds_load_b64
        v2f bb;
        bb.x = load_w(w, kk, j);
        bb.y = load_w(w, kk + 1, j);
        acc2 = __builtin_amdgcn_wmma_f32_16x16x4_f32(
            false, a, false, bb, (short)0, acc2, false, false);
    }

    const float bj = bias[j];                  // biases flat: d*2+o = j
    float* outb = out + (size_t)b * (NNODE * DOUT);
#pragma unroll
    for (int i = 0; i < 8; ++i) {
        const int row = i + 8 * half;
        outb[(m0 + row) * DOUT + j] = acc2[i] + bj;           // coalesced b32 stores
    }
}

extern "C" void kernel_launch(void* const* d_in, const int* in_sizes, int n_in,
                              void* d_out, int out_size, void* d_ws, size_t ws_size,
                              hipStream_t stream) {
    const float* x    = (const float*)d_in[0];   // [128, 400, 2]
    const float* h    = (const float*)d_in[1];   // [128, 400, 128]
    const float* lap  = (const float*)d_in[2];   // [400, 400]
    const float* wgt  = (const float*)d_in[3];   // [65, 64, 2, 2]
    const float* bias = (const float*)d_in[4];   // [64, 2]
    float* out = (float*)d_out;                  // [128, 400*64, 2]

    dim3 grid(NNODE / 16, BATCH);                // 25 x 128 workgroups
    dim3 block(256);                             // 8 wave32 waves
    tgcn_fused_wmma<<<grid, block, 0, stream>>>(x, h, lap, wgt, bias, out);
}